// traj_and_odom_loss_29764123362020
// MI455X (gfx1250) — compile-verified
//
#include <hip/hip_runtime.h>
#include <math.h>

// Problem constants (from reference setup_inputs)
#define BB      64
#define SS      4096
#define CHUNK   64
#define NCHUNK  (SS / CHUNK)      // 64
#define NCHAIN  (2 * BB)          // 128 chains: [0,64) = est, [64,128) = gt
#define NGROUP  (NCHAIN / 4)      // 32 wave-groups of 4 chains for WMMA scan
#define EPSC    1e-6f

typedef __attribute__((ext_vector_type(2))) float v2f;
typedef __attribute__((ext_vector_type(8))) float v8f;

// ---- workspace layout (bytes) ----
// acc doubles: 0:sum sq pos, 1:sum abs pos, 2:sum odom acos,
//              3:sum traj sq, 4:sum traj abs, 5:sum traj acos
#define ACC_DOUBLES 8
#define POSES_OFF   256
#define POSES_BYTES ((size_t)NCHAIN * SS * 12 * 4)            // 25.2 MB (L2-resident)
#define BL_OFF      (POSES_OFF + POSES_BYTES)
#define BL_BYTES    ((size_t)NGROUP * NCHUNK * 32 * 8)        // 512 KB (B-layout cp)
#define PF_OFF      (BL_OFF + BL_BYTES)
#define PF_BYTES    ((size_t)NCHAIN * NCHUNK * 16 * 4)        // 512 KB (16f records)

// ------------------------------------------------------------------ helpers
__device__ __forceinline__ void euler_to_R(float e0, float e1, float e2, float* R) {
  float sa, ca, sb, cb, sc, cc;
  __sincosf(e0, &sa, &ca);
  __sincosf(e1, &sb, &cb);
  __sincosf(e2, &sc, &cc);
  R[0] = cc * cb; R[1] = cc * sb * sa - sc * ca; R[2] = cc * sb * ca + sc * sa;
  R[3] = sc * cb; R[4] = sc * sb * sa + cc * ca; R[5] = sc * sb * ca - cc * sa;
  R[6] = -sb;     R[7] = cb * sa;                R[8] = cb * ca;
}

// C = A @ B for 3x4 affine (implicit bottom row 0,0,0,1), row-major
__device__ __forceinline__ void affmul(const float* __restrict__ A,
                                       const float* __restrict__ Bm,
                                       float* __restrict__ C) {
#pragma unroll
  for (int r = 0; r < 3; ++r) {
#pragma unroll
    for (int c = 0; c < 4; ++c) {
      float acc = (c == 3) ? A[r * 4 + 3] : 0.0f;
      acc = fmaf(A[r * 4 + 0], Bm[0 * 4 + c], acc);
      acc = fmaf(A[r * 4 + 1], Bm[1 * 4 + c], acc);
      acc = fmaf(A[r * 4 + 2], Bm[2 * 4 + c], acc);
      C[r * 4 + c] = acc;
    }
  }
}

__device__ __forceinline__ void load12(const float* __restrict__ p, float* __restrict__ m) {
  const float4* q = (const float4*)p;
  float4 x = q[0], y = q[1], z = q[2];
  m[0] = x.x; m[1] = x.y; m[2]  = x.z; m[3]  = x.w;
  m[4] = y.x; m[5] = y.y; m[6]  = y.z; m[7]  = y.w;
  m[8] = z.x; m[9] = z.y; m[10] = z.z; m[11] = z.w;
}

__device__ __forceinline__ void store12(float* __restrict__ p, const float* __restrict__ m) {
  float4* q = (float4*)p;
  q[0] = make_float4(m[0], m[1], m[2], m[3]);
  q[1] = make_float4(m[4], m[5], m[6], m[7]);
  q[2] = make_float4(m[8], m[9], m[10], m[11]);
}

__device__ __forceinline__ float waveReduce(float v) {
#pragma unroll
  for (int off = 16; off > 0; off >>= 1) v += __shfl_down(v, off, 32);
  return v;
}

// ------------------------------------------------------------------ kernels
__global__ void init_acc(double* acc) {
  if (threadIdx.x < ACC_DOUBLES) acc[threadIdx.x] = 0.0;
}

// Phase 0: build shifted pose matrices (pose[0]->I) + elementwise losses
__global__ void build_poses(const float* __restrict__ yhat,
                            const float* __restrict__ gpos,
                            const float* __restrict__ gori,
                            float* __restrict__ poses,   // [NCHAIN][SS][12]
                            double* __restrict__ acc) {
  int idx = blockIdx.x * blockDim.x + threadIdx.x;   // 0 .. B*S-1
  int b = idx >> 12;
  int s = idx & (SS - 1);
  const float* y  = yhat + (size_t)idx * 6;
  const float* gp = gpos + (size_t)idx * 3;
  const float* go = gori + (size_t)idx * 3;

  float px = y[3], py = y[4], pz = y[5];
  float d0 = px - gp[0], d1 = py - gp[1], d2 = pz - gp[2];
  float sq = d0 * d0 + d1 * d1 + d2 * d2;
  float ab = fabsf(d0) + fabsf(d1) + fabsf(d2);

  float Re[9], Rg[9];
  euler_to_R(y[0], y[1], y[2], Re);
  euler_to_R(go[0], go[1], go[2], Rg);

  float dot = 0.0f;
#pragma unroll
  for (int i = 0; i < 9; ++i) dot = fmaf(Re[i], Rg[i], dot);
  float cosv = fminf(fmaxf((dot - 1.0f) * 0.5f, -1.0f + EPSC), 1.0f - EPSC);
  float ac = acosf(cosv);

  float Pe[12], Pg[12];
  if (s == 0) {
#pragma unroll
    for (int i = 0; i < 12; ++i) { Pe[i] = 0.0f; Pg[i] = 0.0f; }
    Pe[0] = Pe[5] = Pe[10] = 1.0f;
    Pg[0] = Pg[5] = Pg[10] = 1.0f;
  } else {
    Pe[0] = Re[0]; Pe[1] = Re[1]; Pe[2]  = Re[2]; Pe[3]  = px;
    Pe[4] = Re[3]; Pe[5] = Re[4]; Pe[6]  = Re[5]; Pe[7]  = py;
    Pe[8] = Re[6]; Pe[9] = Re[7]; Pe[10] = Re[8]; Pe[11] = pz;
    Pg[0] = Rg[0]; Pg[1] = Rg[1]; Pg[2]  = Rg[2]; Pg[3]  = gp[0];
    Pg[4] = Rg[3]; Pg[5] = Rg[4]; Pg[6]  = Rg[5]; Pg[7]  = gp[1];
    Pg[8] = Rg[6]; Pg[9] = Rg[7]; Pg[10] = Rg[8]; Pg[11] = gp[2];
  }
  store12(poses + ((size_t)b * SS + s) * 12, Pe);
  store12(poses + ((size_t)(BB + b) * SS + s) * 12, Pg);

  sq = waveReduce(sq); ab = waveReduce(ab); ac = waveReduce(ac);
  if ((threadIdx.x & 31) == 0) {
    atomicAdd(&acc[0], (double)sq);
    atomicAdd(&acc[1], (double)ab);
    atomicAdd(&acc[2], (double)ac);
  }
}

// Phase 1: local chunk products; result written DIRECTLY in WMMA B-layout:
// per (group, chunk) a record of 32 lanes x float2, where lane L (n=L&15, h=L>>4)
// holds rows {2h, 2h+1} of the 4x16 B operand at column n; constant bottom row
// (0,0,0,1) is baked in so the scan loop has zero branches.
__global__ void chunk_products(const float* __restrict__ poses,
                               float2* __restrict__ bl) {
  int tid = blockIdx.x * blockDim.x + threadIdx.x;   // 0 .. NCHAIN*NCHUNK-1
  int chain = tid / NCHUNK;
  int chunk = tid % NCHUNK;
  const float* base = poses + ((size_t)chain * SS + (size_t)chunk * CHUNK) * 12;
  float P[12] = {1, 0, 0, 0, 0, 1, 0, 0, 0, 0, 1, 0};
  float M[12], T[12];
  for (int s = 0; s < CHUNK; ++s) {
    __builtin_prefetch(base + (s + 4) * 12, 0, 0);   // global_prefetch_b8
    load12(base + s * 12, M);
    affmul(P, M, T);
#pragma unroll
    for (int i = 0; i < 12; ++i) P[i] = T[i];
  }
  // scatter into B-layout record
  float2* b2 = bl + ((size_t)(chain >> 2) * NCHUNK + chunk) * 32 + (chain & 3) * 4;
#pragma unroll
  for (int col = 0; col < 4; ++col) {
    b2[col]      = make_float2(P[col],     P[4 + col]);                 // rows 0,1
    b2[col + 16] = make_float2(P[8 + col], col == 3 ? 1.0f : 0.0f);     // rows 2,3
  }
}

// Phase 2: per-chain exclusive scan of chunk products.
// One wave per 4 chains; each serial step is ONE v_wmma_f32_16x16x4_f32:
//   A (16x4) = 4 stacked 4x4 prefixes, B (4x16) = 4 side-by-side chunk products,
//   diagonal 4x4 blocks of D = new prefixes, routed through double-buffered LDS
//   (one barrier per step) back into A layout.
__global__ void scan_chunks_wmma(const float2* __restrict__ bl,
                                 float* __restrict__ pf) {   // [NCHAIN][NCHUNK][16]
  __shared__ float lds[128];                // 2 x (4 chains x 16 floats)
  const int lane = threadIdx.x;             // 0..31 (one wave)
  const int n  = lane & 15;
  const int h  = lane >> 4;                 // lane half
  const int iA = n >> 2;                    // chain-in-group for A rows
  const int rA = n & 3;                     // row within 4x4
  const int chain = blockIdx.x * 4 + iA;
  const bool diagOwner = (h == (n >> 3));   // lane holds diag-block elements of D
  const bool hiSel = ((n >> 2) & 1) != 0;   // which half of D vgprs holds them

  // A layout (f32 16x4): lane m+16*(k>>1), vgpr k&1  -> this lane holds
  // P_{iA}[rA][2h] in a.x and P_{iA}[rA][2h+1] in a.y.  Start at identity.
  v2f a;
  a.x = (rA == 2 * h)     ? 1.0f : 0.0f;
  a.y = (rA == 2 * h + 1) ? 1.0f : 0.0f;

  const float2* bsrc = bl + (size_t)blockIdx.x * NCHUNK * 32 + lane;
  float* pdst = pf + (size_t)chain * NCHUNK * 16 + rA * 4 + 2 * h;

  for (int c = 0; c < NCHUNK; ++c) {
    // emit exclusive prefix E[chain][c]: unconditional float2 store (16f record)
    *(float2*)(pdst + (size_t)c * 16) = make_float2(a.x, a.y);

    // B operand: one coalesced b64 load, no branches
    float2 bv2 = bsrc[(size_t)c * 32];
    v2f bv; bv.x = bv2.x; bv.y = bv2.y;

    v8f cz = {};
    v8f d = __builtin_amdgcn_wmma_f32_16x16x4_f32(
        /*neg_a=*/false, a, /*neg_b=*/false, bv,
        /*c_mod=*/(short)0, cz, /*reuse_a=*/false, /*reuse_b=*/false);

    // D element (m,n) lives at vgpr m&7, lane n+16*(m>>3).
    // Diagonal block i = n>>2 occupies rows 4i..4i+3 -> this lane owns them
    // iff h == n>>3; they sit in vgprs base..base+3 with base = ((n>>2)&1)*4.
    float* buf = lds + (c & 1) * 64;
    if (diagOwner) {
      float e0 = hiSel ? d[4] : d[0];
      float e1 = hiSel ? d[5] : d[1];
      float e2 = hiSel ? d[6] : d[2];
      float e3 = hiSel ? d[7] : d[3];
      float* l = buf + (n >> 2) * 16 + (n & 3);
      l[0]  = e0;
      l[4]  = e1;
      l[8]  = e2;
      l[12] = e3;
    }
    __syncthreads();    // double-buffered: single barrier per step is sufficient
    a.x = buf[iA * 16 + rA * 4 + 2 * h];
    a.y = buf[iA * 16 + rA * 4 + 2 * h + 1];
  }
}

// Phase 3: apply prefixes across each chunk, fuse trajectory losses (est vs gt)
__global__ void apply_and_losses(const float* __restrict__ poses,
                                 const float* __restrict__ pf,
                                 double* __restrict__ acc) {
  int tid = blockIdx.x * blockDim.x + threadIdx.x;   // 0 .. BB*NCHUNK-1
  int b = tid / NCHUNK;
  int chunk = tid % NCHUNK;
  int chainE = b, chainG = BB + b;

  float Pe[12], Pg[12], M[12], T[12];
  load12(pf + ((size_t)chainE * NCHUNK + chunk) * 16, Pe);   // rows 0..2 of record
  load12(pf + ((size_t)chainG * NCHUNK + chunk) * 16, Pg);
  const float* pe = poses + ((size_t)chainE * SS + (size_t)chunk * CHUNK) * 12;
  const float* pg = poses + ((size_t)chainG * SS + (size_t)chunk * CHUNK) * 12;

  float tsq = 0.0f, tab = 0.0f, trot = 0.0f;
  for (int s = 0; s < CHUNK; ++s) {
    __builtin_prefetch(pe + (s + 4) * 12, 0, 0);
    __builtin_prefetch(pg + (s + 4) * 12, 0, 0);
    load12(pe + s * 12, M);
    affmul(Pe, M, T);
#pragma unroll
    for (int i = 0; i < 12; ++i) Pe[i] = T[i];
    load12(pg + s * 12, M);
    affmul(Pg, M, T);
#pragma unroll
    for (int i = 0; i < 12; ++i) Pg[i] = T[i];

    float d0 = Pe[3] - Pg[3], d1 = Pe[7] - Pg[7], d2 = Pe[11] - Pg[11];
    tsq += d0 * d0 + d1 * d1 + d2 * d2;
    tab += fabsf(d0) + fabsf(d1) + fabsf(d2);

    float dot = 0.0f;
#pragma unroll
    for (int r = 0; r < 3; ++r)
#pragma unroll
      for (int c2 = 0; c2 < 3; ++c2)
        dot = fmaf(Pe[r * 4 + c2], Pg[r * 4 + c2], dot);
    float cosv = fminf(fmaxf((dot - 1.0f) * 0.5f, -1.0f + EPSC), 1.0f - EPSC);
    trot += acosf(cosv);
  }

  tsq = waveReduce(tsq); tab = waveReduce(tab); trot = waveReduce(trot);
  if ((threadIdx.x & 31) == 0) {
    atomicAdd(&acc[3], (double)tsq);
    atomicAdd(&acc[4], (double)tab);
    atomicAdd(&acc[5], (double)trot);
  }
}

__global__ void finalize(const double* __restrict__ acc, float* __restrict__ out) {
  const double nE = (double)BB * (double)SS;
  const double n3 = nE * 3.0;
  double mse  = acc[0] / n3, mae  = acc[1] / n3, orot = acc[2] / nE;
  double tmse = acc[3] / n3, tmae = acc[4] / n3, trot = acc[5] / nE;
  double odom = orot + (mse + mae);
  double traj = trot + (tmse + tmae);
  out[0] = (float)(0.5 * odom + 0.5 * traj);
}

// ------------------------------------------------------------------ launch
extern "C" void kernel_launch(void* const* d_in, const int* in_sizes, int n_in,
                              void* d_out, int out_size, void* d_ws, size_t ws_size,
                              hipStream_t stream) {
  const float* yhat = (const float*)d_in[0];
  const float* gpos = (const float*)d_in[1];
  const float* gori = (const float*)d_in[2];
  float* out = (float*)d_out;

  char* ws = (char*)d_ws;
  double* acc   = (double*)ws;
  float*  poses = (float*)(ws + POSES_OFF);
  float2* bl    = (float2*)(ws + BL_OFF);
  float*  pf    = (float*)(ws + PF_OFF);

  init_acc<<<1, 32, 0, stream>>>(acc);
  build_poses<<<(BB * SS) / 256, 256, 0, stream>>>(yhat, gpos, gori, poses, acc);
  chunk_products<<<(NCHAIN * NCHUNK) / 256, 256, 0, stream>>>(poses, bl);
  scan_chunks_wmma<<<NGROUP, 32, 0, stream>>>(bl, pf);
  apply_and_losses<<<(BB * NCHUNK) / 256, 256, 0, stream>>>(poses, pf, acc);
  finalize<<<1, 1, 0, stream>>>(acc, out);
}